// SamplerForDream_33689723470498
// MI455X (gfx1250) — compile-verified
//
#include <hip/hip_runtime.h>
#include <hip/hip_bf16.h>
#include <stdint.h>

#define NT 512        // 16 waves of 32
#define NW 16
#define NBIN 256
#define CAND_MAX 1024 // power of two (bitonic)

typedef __attribute__((ext_vector_type(4))) unsigned int u32x4;
typedef __attribute__((ext_vector_type(8))) int          i32x8;
typedef __attribute__((ext_vector_type(4))) int          i32x4;

__device__ __forceinline__ void combine_ms(float& m, float& s, float m2, float s2) {
    float M = fmaxf(m, m2);
    s = s * __expf(m - M) + s2 * __expf(m2 - M);
    m = M;
}

__device__ __forceinline__ void online_upd(float& m, float& s, float& mn, float x) {
    mn = fminf(mn, x);
    if (x > m) { s = s * __expf(m - x) + 1.0f; m = x; }
    else       { s += __expf(x - m); }
}

// One workgroup per row. Row staged global->LDS via TDM (tensor_load_to_lds).
// 3 LDS passes: (1) online softmax stats, (2) 256-bin histogram -> top-k bin,
// (3) collect <=1024 candidates. Bitonic sort, exact top-p/top-k walk on the
// sorted prefix, gumbel gathered only at kept indices.
__global__ void __launch_bounds__(NT)
sampler_row_kernel(const float* __restrict__ logits,
                   const float* __restrict__ gumbel,
                   const float* __restrict__ tempP,
                   const float* __restrict__ toppP,
                   const int*   __restrict__ topkP,
                   float* __restrict__ out,
                   int N, int V)
{
    extern __shared__ __align__(16) float xr[];   // V floats: logits row (125 KB)
    __shared__ int   hist[NW * NBIN];             // 16 KB wave-private histograms
    __shared__ float cval[CAND_MAX];              // 4 KB
    __shared__ int   cidx[CAND_MAX];              // 4 KB
    __shared__ float gsum[CAND_MAX];              // 4 KB  (x + gumbel for kept)
    __shared__ float shm[NW], shs[NW], shn[NW];
    __shared__ float bcast[4];
    __shared__ int   ibcast[4];
    __shared__ int   kcand;

    const int tid  = threadIdx.x;
    const int lane = tid & 31;
    const int wave = tid >> 5;
    const int row  = blockIdx.x;
    const size_t base = (size_t)row * (size_t)V;
    const int V4 = V >> 2;

    const float invT = 1.0f / tempP[0];
    const float topp = toppP[0];
    const int   topk = topkP[0];

    // ---- Stage row into LDS: Tensor Data Mover (CDNA5) ----
#if __has_builtin(__builtin_amdgcn_tensor_load_to_lds)
    if (tid < 32) {  // wave 0 issues the DMA, waits TENSORcnt; barrier releases the rest
        unsigned lds_off = (unsigned)(uintptr_t)(xr);
        unsigned long long ga = (unsigned long long)(uintptr_t)(logits + base);
        u32x4 g0;
        g0.x = 1u;                                                    // count=1
        g0.y = lds_off;                                               // lds_addr
        g0.z = (unsigned)(ga & 0xFFFFFFFFull);                        // global_addr[31:0]
        g0.w = (unsigned)((ga >> 32) & 0x01FFFFFFull) | (2u << 30);   // addr[56:32] | type=2
        i32x8 g1;
        g1.s0 = (int)(2u << 16);                                      // data_size = 4B
        g1.s1 = (int)(((unsigned)V & 0xFFFFu) << 16);                 // tensor_dim0 lo
        g1.s2 = (int)((((unsigned)V >> 16) & 0xFFFFu) | (1u << 16));  // tensor_dim0 hi | tensor_dim1=1
        g1.s3 = (int)(((unsigned)V & 0xFFFFu) << 16);                 // tile_dim0 = V
        g1.s4 = 1;                                                    // tile_dim1 = 1
        g1.s5 = V;                                                    // tensor_dim0_stride
        g1.s6 = 0; g1.s7 = 0;
        i32x4 g2 = {0, 0, 0, 0};
        i32x4 g3 = {0, 0, 0, 0};
#if defined(__clang_major__) && (__clang_major__ >= 23)
        i32x8 gz = {0, 0, 0, 0, 0, 0, 0, 0};
        __builtin_amdgcn_tensor_load_to_lds(g0, g1, g2, g3, gz, 0);
#else
        __builtin_amdgcn_tensor_load_to_lds(g0, g1, g2, g3, 0);
#endif
#if __has_builtin(__builtin_amdgcn_s_wait_tensorcnt)
        __builtin_amdgcn_s_wait_tensorcnt(0);
#else
        asm volatile("s_wait_tensorcnt 0x0" ::: "memory");
#endif
    }
#else
    for (int i = tid; i < V; i += NT) xr[i] = logits[base + i];
#endif
    __syncthreads();

    const float4* xr4 = (const float4*)xr;

    // ---- Pass 1: online softmax (M, Z_all) + row min, b128 LDS reads ----
    float m = -3.0e38f, s = 0.0f, mn = 3.0e38f;
    for (int i = tid; i < V4; i += NT) {
        float4 v = xr4[i];
        online_upd(m, s, mn, v.x * invT);
        online_upd(m, s, mn, v.y * invT);
        online_upd(m, s, mn, v.z * invT);
        online_upd(m, s, mn, v.w * invT);
    }
    for (int i = (V4 << 2) + tid; i < V; i += NT) online_upd(m, s, mn, xr[i] * invT);

    for (int off = 16; off >= 1; off >>= 1) {
        float m2 = __shfl_xor(m, off, 32);
        float s2 = __shfl_xor(s, off, 32);
        float n2 = __shfl_xor(mn, off, 32);
        combine_ms(m, s, m2, s2);
        mn = fminf(mn, n2);
    }
    if (lane == 0) { shm[wave] = m; shs[wave] = s; shn[wave] = mn; }
    __syncthreads();
    if (tid == 0) {
        float M = shm[0], S = shs[0], Nn = shn[0];
        for (int w = 1; w < NW; ++w) { combine_ms(M, S, shm[w], shs[w]); Nn = fminf(Nn, shn[w]); }
        bcast[0] = M; bcast[1] = S; bcast[2] = Nn;
    }
    __syncthreads();
    const float M       = bcast[0];
    const float Zall    = bcast[1];
    const float rowmin  = bcast[2];
    const float pthresh = topp * Zall;

    // ---- Pass 2 (+rare refinements): histogram-select the top-k cutoff bin ----
    float rlo = rowmin - 1.0f;   // all tokens lie in (rlo, M]
    float rhi = M;
    int   nAbove = 0;
    int   totalCand = 0;
    for (int ref = 0; ref < 3; ++ref) {
        for (int b = tid; b < NW * NBIN; b += NT) hist[b] = 0;
        __syncthreads();
        const float invw = (float)NBIN / (rhi - rlo);
        const int hb = wave << 8;
        for (int i = tid; i < V4; i += NT) {
            float4 v = xr4[i];
            float e0 = v.x * invT, e1 = v.y * invT, e2 = v.z * invT, e3 = v.w * invT;
            if (e0 > rlo && e0 <= rhi) atomicAdd(&hist[hb + min(NBIN - 1, max(0, (int)((e0 - rlo) * invw)))], 1);
            if (e1 > rlo && e1 <= rhi) atomicAdd(&hist[hb + min(NBIN - 1, max(0, (int)((e1 - rlo) * invw)))], 1);
            if (e2 > rlo && e2 <= rhi) atomicAdd(&hist[hb + min(NBIN - 1, max(0, (int)((e2 - rlo) * invw)))], 1);
            if (e3 > rlo && e3 <= rhi) atomicAdd(&hist[hb + min(NBIN - 1, max(0, (int)((e3 - rlo) * invw)))], 1);
        }
        for (int i = (V4 << 2) + tid; i < V; i += NT) {
            float e = xr[i] * invT;
            if (e > rlo && e <= rhi) atomicAdd(&hist[hb + min(NBIN - 1, max(0, (int)((e - rlo) * invw)))], 1);
        }
        __syncthreads();
        if (tid < NBIN) {
            int t = 0;
            for (int w = 0; w < NW; ++w) t += hist[(w << 8) + tid];
            hist[tid] = t;
        }
        __syncthreads();
        if (tid == 0) {
            int cum = nAbove, b = NBIN - 1;
            for (; b >= 0; --b) { if (cum + hist[b] >= topk) break; cum += hist[b]; }
            if (b < 0) b = 0;
            float w = (rhi - rlo) * (1.0f / (float)NBIN);
            float binlo = rlo + w * (float)b;
            float binhi = (b == NBIN - 1) ? rhi : (rlo + w * (float)(b + 1));
            bcast[0] = binlo; bcast[1] = binhi;
            ibcast[0] = cum;               // tokens strictly above binhi
            ibcast[1] = cum + hist[b];     // candidates (tokens > binlo)
        }
        __syncthreads();
        totalCand = ibcast[1];
        float binlo = bcast[0], binhi = bcast[1];
        if (totalCand <= CAND_MAX || !(binhi > binlo)) { rlo = binlo; break; }
        nAbove = ibcast[0]; rlo = binlo; rhi = binhi;   // refine inside the bin
    }
    const float t0 = rlo;   // candidates = { x > t0 } (superset of kept set)

    // ---- Pass 3: collect candidates (value, index) into LDS ----
    if (tid == 0) kcand = 0;
    __syncthreads();
    for (int i = tid; i < V4; i += NT) {
        float4 v = xr4[i];
        float e0 = v.x * invT, e1 = v.y * invT, e2 = v.z * invT, e3 = v.w * invT;
        int ib = i << 2;
        if (e0 > t0) { int p = atomicAdd(&kcand, 1); if (p < CAND_MAX) { cval[p] = e0; cidx[p] = ib; } }
        if (e1 > t0) { int p = atomicAdd(&kcand, 1); if (p < CAND_MAX) { cval[p] = e1; cidx[p] = ib + 1; } }
        if (e2 > t0) { int p = atomicAdd(&kcand, 1); if (p < CAND_MAX) { cval[p] = e2; cidx[p] = ib + 2; } }
        if (e3 > t0) { int p = atomicAdd(&kcand, 1); if (p < CAND_MAX) { cval[p] = e3; cidx[p] = ib + 3; } }
    }
    for (int i = (V4 << 2) + tid; i < V; i += NT) {
        float e = xr[i] * invT;
        if (e > t0) { int p = atomicAdd(&kcand, 1); if (p < CAND_MAX) { cval[p] = e; cidx[p] = i; } }
    }
    __syncthreads();
    int nc = kcand; if (nc > CAND_MAX) nc = CAND_MAX;

    // pad & bitonic sort descending (index-ascending tie break => deterministic)
    for (int i = tid; i < CAND_MAX; i += NT)
        if (i >= nc) { cval[i] = -3.0e38f; cidx[i] = 0x7FFFFFFF; }
    __syncthreads();
    for (int ksz = 2; ksz <= CAND_MAX; ksz <<= 1) {
        for (int jsz = ksz >> 1; jsz >= 1; jsz >>= 1) {
            for (int i = tid; i < CAND_MAX; i += NT) {
                int ixj = i ^ jsz;
                if (ixj > i) {
                    bool desc = ((i & ksz) == 0);
                    float a = cval[i], b2 = cval[ixj];
                    int   ai = cidx[i], bi2 = cidx[ixj];
                    bool swap;
                    if (a == b2) swap = desc ? (ai > bi2) : (ai < bi2);
                    else         swap = desc ? (a < b2)   : (a > b2);
                    if (swap) { cval[i] = b2; cval[ixj] = a; cidx[i] = bi2; cidx[ixj] = ai; }
                }
            }
            __syncthreads();
        }
    }

    // ---- Exact top-p/top-k walk on sorted prefix: j kept tokens, Z_kept ----
    if (tid == 0) {
        float cum = 0.0f; int j = 0;
        for (int q = 0; q < nc; ++q) {
            if (q >= topk) break;            // top-k: at most k tokens
            if (cum > pthresh) break;        // top-p: exclusive cum > p -> removed
            cum += __expf(cval[q] - M);
            ++j;
        }
        ibcast[0] = j;
        bcast[0]  = cum;                     // Z_kept
    }
    __syncthreads();
    const int   j  = ibcast[0];
    const float Zk = bcast[0];

    // ---- Gumbel gather at kept indices only, argmax sample ----
    if (tid < j && tid < NT) gsum[tid] = cval[tid] + gumbel[base + cidx[tid]];
    __syncthreads();
    if (tid == 0) {
        float best = -3.0e38f; int bq = 0;
        int jj = j < NT ? j : NT;
        for (int q = 0; q < jj; ++q) {
            if (gsum[q] > best) { best = gsum[q]; bq = q; }
        }
        float conf = __expf(cval[bq] - M) / Zk;
        out[row]         = conf;             // confidence
        out[N + row]     = (float)cidx[bq];  // sampled token x0
        out[2 * N + row] = conf;             // initial_confidence
    }
}

// Cross-row accept: any(conf > thr) ? (conf > thr) : one-hot(argmax(conf))
__global__ void __launch_bounds__(256)
accept_kernel(const float* __restrict__ conf,
              const float* __restrict__ thrP,
              float* __restrict__ acc, int N)
{
    __shared__ float sv[8];
    __shared__ int   si[8];
    __shared__ int   sa[8];
    const float thr = thrP[0];
    const int tid = threadIdx.x;
    const int lane = tid & 31, wave = tid >> 5;

    float bestv = -3.0e38f; int besti = 0; int anyl = 0;
    for (int r = tid; r < N; r += 256) {
        float c = conf[r];
        if (c > thr) anyl = 1;
        if (c > bestv || (c == bestv && r < besti)) { bestv = c; besti = r; }
    }
    for (int off = 16; off >= 1; off >>= 1) {
        float v2 = __shfl_xor(bestv, off, 32);
        int   i2 = __shfl_xor(besti, off, 32);
        int   a2 = __shfl_xor(anyl,  off, 32);
        anyl |= a2;
        if (v2 > bestv || (v2 == bestv && i2 < besti)) { bestv = v2; besti = i2; }
    }
    if (lane == 0) { sv[wave] = bestv; si[wave] = besti; sa[wave] = anyl; }
    __syncthreads();
    if (tid == 0) {
        for (int w = 1; w < 8; ++w) {
            anyl |= sa[w];
            if (sv[w] > bestv || (sv[w] == bestv && si[w] < besti)) { bestv = sv[w]; besti = si[w]; }
        }
        sa[0] = anyl; si[0] = besti;
    }
    __syncthreads();
    const int any = sa[0];
    const int amax = si[0];
    for (int r = tid; r < N; r += 256) {
        float c = conf[r];
        acc[r] = any ? (c > thr ? 1.0f : 0.0f) : (r == amax ? 1.0f : 0.0f);
    }
}

extern "C" void kernel_launch(void* const* d_in, const int* in_sizes, int n_in,
                              void* d_out, int out_size, void* d_ws, size_t ws_size,
                              hipStream_t stream) {
    const float* logits = (const float*)d_in[0];
    const float* gumbel = (const float*)d_in[1];
    const float* tempP  = (const float*)d_in[2];
    const float* toppP  = (const float*)d_in[3];
    const int*   topkP  = (const int*)d_in[4];
    const float* thrP   = (const float*)d_in[5];
    float* out = (float*)d_out;

    const int N = out_size / 4;          // 4 concatenated [N] outputs
    const int V = in_sizes[0] / N;       // 32000
    size_t shmem = (size_t)V * sizeof(float);   // 125 KB row in LDS; +~29 KB static

    sampler_row_kernel<<<N, NT, shmem, stream>>>(logits, gumbel, tempP, toppP, topkP, out, N, V);
    accept_kernel<<<1, 256, 0, stream>>>(out, thrP, out + 3 * (size_t)N, N);
}